// VoronoiTessellation_63548336111938
// MI455X (gfx1250) — compile-verified
//
#include <hip/hip_runtime.h>
#include <math.h>

typedef __attribute__((ext_vector_type(2))) float v2f;
typedef __attribute__((ext_vector_type(8))) float v8f;

#define ALPHA_C 0.01f
#define EPSB_C  (ALPHA_C / (1.0f - 2.0f * ALPHA_C))
#define BIG_C   1e9f
#define KANCH   1024
#define DDIM    16
#define LDA     17   // padded LDS row stride (coprime with 64 banks)

// workspace layout (floats):
//   [0, K*16)              transformed anchors a[k][d]
//   [K*16, K*17)           a_sq[k]
//   [K*17, K*18)           logprobs[k]  (log_softmax of mixture logits)
//   [K*18]                 Cg (scalar logdet constant)

__global__ __launch_bounds__(1024)
void vor_prep_kernel(const float* __restrict__ anchor_raw,
                     const float* __restrict__ box_min,
                     const float* __restrict__ box_max,
                     const float* __restrict__ an1_w,
                     const float* __restrict__ shift_coeff,
                     const float* __restrict__ an2_w,
                     const float* __restrict__ mix_logits,
                     float* __restrict__ ws)
{
    __shared__ float red[1024];
    const int t = threadIdx.x;
    float* a_ws   = ws;
    float* asq_ws = ws + KANCH * DDIM;
    float* lp_ws  = asq_ws + KANCH;
    float* cg_ws  = lp_ws + KANCH;

    // anchor transform: R -> (-1,1) -> (0,1) -> box, plus squared norm
    float sq = 0.f;
    for (int d = 0; d < DDIM; ++d) {
        float raw = anchor_raw[t * DDIM + d];
        float aa  = raw / (1.0f + fabsf(raw));
        float bmn = box_min[d];
        float sc  = box_max[d] - bmn;
        float val = (aa + 1.0f) * 0.5f * sc + bmn;
        a_ws[t * DDIM + d] = val;
        sq += val * val;
    }
    asq_ws[t] = sq;

    // log_softmax over mixture logits (K == blockDim == 1024)
    float m = mix_logits[t];
    red[t] = m;
    __syncthreads();
    for (int s = 512; s > 0; s >>= 1) {
        if (t < s) red[t] = fmaxf(red[t], red[t + s]);
        __syncthreads();
    }
    float mx = red[0];
    __syncthreads();
    red[t] = __expf(m - mx);
    __syncthreads();
    for (int s = 512; s > 0; s >>= 1) {
        if (t < s) red[t] += red[t + s];
        __syncthreads();
    }
    float lse = mx + __logf(red[0]);
    lp_ws[t] = m - lse;

    if (t == 0) {
        float cg = 0.f;
        const float span = 1.0f + 2.0f * EPSB_C;
        for (int d = 0; d < DDIM; ++d) {
            cg -= an1_w[d];
            cg -= an2_w[d];
            float sc = box_max[d] - box_min[d];
            cg -= __logf(sc);
            float c = 0.98f / (1.0f + __expf(-shift_coeff[d])) + 0.01f;
            cg -= log1pf(-c);
        }
        cg += DDIM * __logf(1.0f - 2.0f * ALPHA_C);  // -ld constant part
        cg += DDIM * __logf(span);
        cg_ws[0] = cg;
    }
}

__global__ __launch_bounds__(256)
void vor_main_kernel(const float* __restrict__ x_in,
                     const float* __restrict__ logp_in,
                     const float* __restrict__ an1_b,
                     const float* __restrict__ an1_w,
                     const float* __restrict__ an2_b,
                     const float* __restrict__ an2_w,
                     const float* __restrict__ shift_coeff,
                     const float* __restrict__ box_min,
                     const float* __restrict__ box_max,
                     const float* __restrict__ ws,
                     float* __restrict__ out_x,
                     float* __restrict__ out_logp,
                     float* __restrict__ out_mask)
{
    extern __shared__ float smem[];
    float* a_lds   = smem;                        // 1024*17
    float* asq_lds = smem + KANCH * LDA;          // 1024
    float* cds     = asq_lds + KANCH;             // 16*8 per-dim constants
    int*   klds    = (int*)(cds + DDIM * 8);      // 8 waves * 16 rows

    const float* a_ws   = ws;
    const float* asq_ws = ws + KANCH * DDIM;
    const float* lp_ws  = asq_ws + KANCH;

    const int tid  = threadIdx.x;
    const int lane = tid & 31;
    const int wv   = tid >> 5;

    // ---- stage per-dim constants ----
    if (tid < DDIM) {
        int d = tid;
        cds[d * 8 + 0] = an1_b[d];
        cds[d * 8 + 1] = __expf(an1_w[d]);
        float bmn = box_min[d], bmx = box_max[d];
        cds[d * 8 + 2] = bmx - bmn;
        cds[d * 8 + 3] = bmn;
        cds[d * 8 + 4] = bmx;
        float c = 0.98f / (1.0f + __expf(-shift_coeff[d])) + 0.01f;
        cds[d * 8 + 5] = 1.0f - c;
        cds[d * 8 + 6] = an2_b[d];
        cds[d * 8 + 7] = __expf(an2_w[d]);
    }
    // ---- stage anchors + a_sq into LDS (padded rows) ----
    for (int i4 = tid; i4 < KANCH * 4; i4 += 256) {
        int k = i4 >> 2, db = (i4 & 3) * 4;
        float4 v = ((const float4*)a_ws)[i4];
        a_lds[k * LDA + db + 0] = v.x;
        a_lds[k * LDA + db + 1] = v.y;
        a_lds[k * LDA + db + 2] = v.z;
        a_lds[k * LDA + db + 3] = v.w;
    }
    for (int i = tid; i < KANCH; i += 256) asq_lds[i] = asq_ws[i];
    __syncthreads();

    const int rowbase = (blockIdx.x * 8 + wv) * 16;
    const int myrow   = rowbase + (lane & 15);
    const int koff    = (lane >> 4) * 2;   // 0 for lanes 0-15, 2 for 16-31 (A layout)

    // ---- phase 0: actnorm1 + logit-inverse + box scale, in WMMA A-layout ----
    v2f z[4];
    float sls = 0.f;
    #pragma unroll
    for (int c = 0; c < 4; ++c) {
        int d0 = 4 * c + koff;
        float2 xv = *(const float2*)(x_in + (size_t)myrow * 16 + d0);
        float xx[2] = {xv.x, xv.y};
        float zz[2];
        #pragma unroll
        for (int e = 0; e < 2; ++e) {
            int d = d0 + e;
            float y  = (xx[e] + cds[d * 8 + 0]) * cds[d * 8 + 1];
            float ay = fabsf(y);
            sls += -(ay + 2.0f * log1pf(__expf(-ay)));   // logsig(y)+logsig(-y)
            float sig = 1.0f / (1.0f + __expf(-y));
            float tv  = (sig - ALPHA_C) * (1.0f / (1.0f - 2.0f * ALPHA_C));
            tv = (tv + EPSB_C) * (1.0f / (1.0f + 2.0f * EPSB_C));
            zz[e] = tv * cds[d * 8 + 2] + cds[d * 8 + 3];
        }
        z[c].x = zz[0]; z[c].y = zz[1];
    }
    sls += __shfl_xor(sls, 16, 32);

    // ---- phase 1: nearest anchor via WMMA, acc = x.aT - 0.5*a_sq (maximize) ----
    float bestv[8]; int bestk[8];
    #pragma unroll
    for (int j = 0; j < 8; ++j) { bestv[j] = -BIG_C; bestk[j] = 0; }

    #pragma unroll 2
    for (int n0 = 0; n0 < KANCH; n0 += 16) {
        int n = n0 + (lane & 15);
        float cinit = -0.5f * asq_lds[n];
        v8f acc;
        #pragma unroll
        for (int j = 0; j < 8; ++j) acc[j] = cinit;
        #pragma unroll
        for (int c = 0; c < 4; ++c) {
            const float* bp = &a_lds[n * LDA + 4 * c + koff];
            v2f bv; bv.x = bp[0]; bv.y = bp[1];
            acc = __builtin_amdgcn_wmma_f32_16x16x4_f32(
                false, z[c], false, bv, (short)0, acc, false, false);
        }
        #pragma unroll
        for (int j = 0; j < 8; ++j) {
            bool take = acc[j] > bestv[j];
            bestv[j] = take ? acc[j] : bestv[j];
            bestk[j] = take ? n : bestk[j];
        }
    }
    // branchless argmax reduce across 16-lane halves (masks 1..8 keep halves apart)
    #pragma unroll
    for (int m = 1; m <= 8; m <<= 1) {
        #pragma unroll
        for (int j = 0; j < 8; ++j) {
            float ov = __shfl_xor(bestv[j], m, 32);
            int   ok = __shfl_xor(bestk[j], m, 32);
            bool take = (ov > bestv[j]) | ((ov == bestv[j]) & (ok < bestk[j]));
            bestv[j] = take ? ov : bestv[j];
            bestk[j] = take ? ok : bestk[j];
        }
    }
    // redistribute k to per-row layout: row M = lane&15
    const int idx  = lane & 7;
    const int srcl = ((lane & 8) << 1) | (lane & 7);
    int ksel;
    {
        int t01 = (idx & 1) ? bestk[1] : bestk[0];
        int t23 = (idx & 1) ? bestk[3] : bestk[2];
        int t45 = (idx & 1) ? bestk[5] : bestk[4];
        int t67 = (idx & 1) ? bestk[7] : bestk[6];
        int t03 = (idx & 2) ? t23 : t01;
        int t47 = (idx & 2) ? t67 : t45;
        ksel    = (idx & 4) ? t47 : t03;
    }
    const int krow = __shfl(ksel, srcl, 32);

    // ---- shift toward anchor, dx, r, u, box distances (all A-layout) ----
    v2f xk[4], dxv[4], uu[4];
    float r2p = 0.f;
    #pragma unroll
    for (int c = 0; c < 4; ++c) {
        int d0 = 4 * c + koff;
        float k0 = a_lds[krow * LDA + d0];
        float k1 = a_lds[krow * LDA + d0 + 1];
        xk[c].x = k0; xk[c].y = k1;
        dxv[c].x = cds[d0 * 8 + 5]       * (z[c].x - k0);   // (1-c)*(z - xk)
        dxv[c].y = cds[(d0 + 1) * 8 + 5] * (z[c].y - k1);
        r2p += dxv[c].x * dxv[c].x + dxv[c].y * dxv[c].y;
    }
    float r2   = r2p + __shfl_xor(r2p, 16, 32);
    float r    = sqrtf(r2);
    float rinv = 1.0f / fmaxf(r, 1e-12f);

    float udxkp = 0.f, tboxp = BIG_C;
    #pragma unroll
    for (int c = 0; c < 4; ++c) {
        uu[c].x = dxv[c].x * rinv;
        uu[c].y = dxv[c].y * rinv;
        udxkp += uu[c].x * xk[c].x + uu[c].y * xk[c].y;
        int d0 = 4 * c + koff;
        #pragma unroll
        for (int e = 0; e < 2; ++e) {
            float u  = e ? uu[c].y : uu[c].x;
            float kk = e ? xk[c].y : xk[c].x;
            int d = d0 + e;
            float sel = (u > 0.f) ? cds[d * 8 + 4] : cds[d * 8 + 3];
            float tb  = (fabsf(u) > 1e-12f) ? (sel - kk) / u : BIG_C;
            tboxp = fminf(tboxp, tb);
        }
    }
    float udxk = udxkp + __shfl_xor(udxkp, 16, 32);
    float tbox = fminf(tboxp, __shfl_xor(tboxp, 16, 32));
    float asqk = asq_lds[krow];

    // broadcast per-row scalars into per-j (C/D row) layout
    float udxk_b[8], asqk_b[8]; int k_b[8];
    const int hi8 = (lane & 16) ? 8 : 0;
    #pragma unroll
    for (int j = 0; j < 8; ++j) {
        int s = j + hi8;
        udxk_b[j] = __shfl(udxk, s, 32);
        asqk_b[j] = __shfl(asqk, s, 32);
        k_b[j]    = __shfl(krow, s, 32);
    }

    // ---- phase 2: half-plane distances via two WMMA chains (u.aT, a_k.aT) ----
    // Track argmin of h/p as (h,p) pair: h1/p1 < h2/p2  <=>  h1*p2 < h2*p1  (p > 0).
    float hmin[8], pmin[8];
    #pragma unroll
    for (int j = 0; j < 8; ++j) { hmin[j] = BIG_C; pmin[j] = 1.0f; }

    for (int n0 = 0; n0 < KANCH; n0 += 16) {
        int n = n0 + (lane & 15);
        v8f accU = {0.f,0.f,0.f,0.f,0.f,0.f,0.f,0.f};
        v8f accG = {0.f,0.f,0.f,0.f,0.f,0.f,0.f,0.f};
        #pragma unroll
        for (int c = 0; c < 4; ++c) {
            const float* bp = &a_lds[n * LDA + 4 * c + koff];
            v2f bv; bv.x = bp[0]; bv.y = bp[1];
            accU = __builtin_amdgcn_wmma_f32_16x16x4_f32(
                false, uu[c], false, bv, (short)0, accU, false, false);
            accG = __builtin_amdgcn_wmma_f32_16x16x4_f32(
                false, xk[c], false, bv, (short)0, accG, false, false);
        }
        float asqn = asq_lds[n];
        #pragma unroll
        for (int j = 0; j < 8; ++j) {
            float p  = accU[j] - udxk_b[j];
            float h  = 0.5f * (asqn + asqk_b[j]) - accG[j];
            bool valid = (n != k_b[j]) & (p > 1e-12f);
            float hv = valid ? h : BIG_C;
            float pv = valid ? p : 1.0f;
            bool take = hv * pmin[j] < hmin[j] * pv;
            hmin[j] = take ? hv : hmin[j];
            pmin[j] = take ? pv : pmin[j];
        }
    }
    #pragma unroll
    for (int m = 1; m <= 8; m <<= 1) {
        #pragma unroll
        for (int j = 0; j < 8; ++j) {
            float oh = __shfl_xor(hmin[j], m, 32);
            float op = __shfl_xor(pmin[j], m, 32);
            bool take = oh * pmin[j] < hmin[j] * op;
            hmin[j] = take ? oh : hmin[j];
            pmin[j] = take ? op : pmin[j];
        }
    }
    // redistribute (h,p) to per-row layout and do the single division
    float hsel, psel;
    {
        float h01 = (idx & 1) ? hmin[1] : hmin[0];
        float h23 = (idx & 1) ? hmin[3] : hmin[2];
        float h45 = (idx & 1) ? hmin[5] : hmin[4];
        float h67 = (idx & 1) ? hmin[7] : hmin[6];
        float h03 = (idx & 2) ? h23 : h01;
        float h47 = (idx & 2) ? h67 : h45;
        hsel      = (idx & 4) ? h47 : h03;
        float p01 = (idx & 1) ? pmin[1] : pmin[0];
        float p23 = (idx & 1) ? pmin[3] : pmin[2];
        float p45 = (idx & 1) ? pmin[5] : pmin[4];
        float p67 = (idx & 1) ? pmin[7] : pmin[6];
        float p03 = (idx & 2) ? p23 : p01;
        float p47 = (idx & 2) ? p67 : p45;
        psel      = (idx & 4) ? p47 : p03;
    }
    float thalf = __shfl(hsel, srcl, 32) / __shfl(psel, srcl, 32);
    float tdist = fminf(thalf, tbox);

    // ---- radial map + actnorm2 + outputs ----
    float ratio = r / tdist;
    float sfac  = 1.0f / (1.0f - ratio);
    #pragma unroll
    for (int c = 0; c < 4; ++c) {
        int d0 = 4 * c + koff;
        float2 ov;
        ov.x = (dxv[c].x * sfac + cds[d0 * 8 + 6])       * cds[d0 * 8 + 7];
        ov.y = (dxv[c].y * sfac + cds[(d0 + 1) * 8 + 6]) * cds[(d0 + 1) * 8 + 7];
        *(float2*)(out_x + (size_t)myrow * 16 + d0) = ov;
    }
    if (lane < 16) {
        float Cg = lp_ws[KANCH];  // scalar constant after logprobs
        float lg = logp_in[myrow] + Cg - sls
                 + (DDIM + 1.0f) * log1pf(-ratio) - lp_ws[krow];
        out_logp[myrow] = lg;
        klds[wv * 16 + lane] = krow;
    }
    __syncthreads();

    // ---- one-hot mask (dominant HBM traffic; coalesced float4 stores) ----
    for (int rr = 0; rr < 16; ++rr) {
        int row = rowbase + rr;
        int kk  = klds[wv * 16 + rr];
        for (int col = lane * 4; col < KANCH; col += 128) {
            float4 mv;
            mv.x = (col + 0 == kk) ? 1.0f : 0.0f;
            mv.y = (col + 1 == kk) ? 1.0f : 0.0f;
            mv.z = (col + 2 == kk) ? 1.0f : 0.0f;
            mv.w = (col + 3 == kk) ? 1.0f : 0.0f;
            *(float4*)(out_mask + (size_t)row * KANCH + col) = mv;
        }
    }
}

extern "C" void kernel_launch(void* const* d_in, const int* in_sizes, int n_in,
                              void* d_out, int out_size, void* d_ws, size_t ws_size,
                              hipStream_t stream) {
    const float* x        = (const float*)d_in[0];
    const float* logp     = (const float*)d_in[1];
    const float* an1_w    = (const float*)d_in[2];
    const float* an1_b    = (const float*)d_in[3];
    const float* an2_w    = (const float*)d_in[4];
    const float* an2_b    = (const float*)d_in[5];
    const float* shift_c  = (const float*)d_in[6];
    const float* anchor   = (const float*)d_in[7];
    const float* box_min  = (const float*)d_in[8];
    const float* box_max  = (const float*)d_in[9];
    const float* mix      = (const float*)d_in[10];

    const int B = in_sizes[0] / DDIM;   // 32768
    float* ws = (float*)d_ws;
    float* out_x    = (float*)d_out;
    float* out_logp = out_x + (size_t)B * DDIM;
    float* out_mask = out_logp + B;

    // shared mem: anchors (padded) + a_sq + consts + klds
    const size_t shmem = (KANCH * LDA + KANCH + DDIM * 8 + 8 * 16) * sizeof(float);
    (void)hipFuncSetAttribute((const void*)vor_main_kernel,
                              hipFuncAttributeMaxDynamicSharedMemorySize, (int)shmem);

    vor_prep_kernel<<<1, 1024, 0, stream>>>(anchor, box_min, box_max,
                                            an1_w, shift_c, an2_w, mix, ws);

    vor_main_kernel<<<B / 128, 256, shmem, stream>>>(
        x, logp, an1_b, an1_w, an2_b, an2_w, shift_c,
        box_min, box_max, ws, out_x, out_logp, out_mask);
}